// VariationalAggregator_4458176053677
// MI455X (gfx1250) — compile-verified
//
#include <hip/hip_runtime.h>
#include <hip/hip_bf16.h>
#include <math.h>

#define B_    64
#define S_    2048
#define D_    128
#define NBINS 48
#define TPER  48.0f
#define WROW  256        // mu(128) + logvar(128)
#define TOKMASK 0x1FFFFFu
#define BINSHIFT 21
#define BININVALID 63u

typedef __attribute__((ext_vector_type(2))) float v2f;
typedef __attribute__((ext_vector_type(8))) float v8f;

// ---------------- deterministic counter-based N(0,1) ----------------
__device__ __forceinline__ unsigned mix32(unsigned x) {
  x ^= x >> 16; x *= 0x7feb352dU;
  x ^= x >> 15; x *= 0x846ca68bU;
  x ^= x >> 16;
  return x;
}
__device__ __forceinline__ float hash_gauss(unsigned idx) {
  unsigned h1 = mix32(idx * 2654435761u + 0x9E3779B9u);
  unsigned h2 = mix32(idx * 2246822519u + 0x85EBCA77u);
  float u1 = (float)(h1 >> 8) * (1.0f / 16777216.0f);
  float u2 = (float)(h2 >> 8) * (1.0f / 16777216.0f);
  u1 = fmaxf(u1, 1.0e-12f);
  return sqrtf(-2.0f * __logf(u1)) * __cosf(6.28318530718f * u2);
}

// ---------------- kernel 1: per-(b,s) entropy H + packed token ----------------
__global__ void k1_entropy(const float* __restrict__ X, const float* __restrict__ W,
                           float* __restrict__ H, unsigned* __restrict__ tokbin,
                           float* __restrict__ kl_slot) {
  int idx = blockIdx.x * blockDim.x + threadIdx.x;
  if (idx == 0) *kl_slot = 0.0f;           // zero KL accumulator each launch
  if (idx >= B_ * S_) return;
  float T  = X[(size_t)idx * 2 + 0];
  int  tok = (int)X[(size_t)idx * 2 + 1];
  tokbin[idx] = (unsigned)tok & TOKMASK;   // bin bits filled by kernel 2
  const float4* lv = (const float4*)(W + (size_t)tok * WROW + D_);
  float s = 0.f;
#pragma unroll 8
  for (int j = 0; j < D_ / 4; ++j) { float4 v = lv[j]; s += v.x + v.y + v.z + v.w; }
  const float LOG2PI = 1.8378770664093453f;
  float mask = (T < TPER) ? 1.0f : 0.0f;
  H[idx] = 0.5f * ((float)D_ + (float)D_ * LOG2PI + s) * mask;
}

// ---------------- kernel 2: block scan -> normalized cumsum -> pack bins ----------------
__global__ void k2_scan_bins(const float* __restrict__ H, unsigned* __restrict__ tokbin) {
  const int PER = S_ / 256;  // 8 elements per thread
  int b = blockIdx.x, tid = threadIdx.x;
  int base = b * S_ + tid * PER;

  float loc[PER]; float s = 0.f;
#pragma unroll
  for (int j = 0; j < PER; ++j) { loc[j] = H[base + j]; s += loc[j]; }

  __shared__ float sc[256];
  sc[tid] = s; __syncthreads();
  for (int off = 1; off < 256; off <<= 1) {     // Hillis-Steele inclusive scan
    float v = (tid >= off) ? sc[tid - off] : 0.f;
    __syncthreads();
    sc[tid] += v;
    __syncthreads();
  }
  float prefix = sc[tid] - s;                    // exclusive prefix for this thread

  float run = prefix, mymax = -3.4e38f;
  float cum[PER];
#pragma unroll
  for (int j = 0; j < PER; ++j) { run += loc[j]; cum[j] = run; mymax = fmaxf(mymax, run); }

  __shared__ float mx[256];
  mx[tid] = mymax; __syncthreads();
  for (int off = 128; off > 0; off >>= 1) {
    if (tid < off) mx[tid] = fmaxf(mx[tid], mx[tid + off]);
    __syncthreads();
  }
  float gmax = mx[0];

#pragma unroll
  for (int j = 0; j < PER; ++j) {
    float hc = cum[j] / gmax;
    int bn = (int)floorf(hc * (float)NBINS);
    unsigned b6 = (hc >= 0.f && hc < 1.f) ? (unsigned)bn : BININVALID;
    tokbin[base + j] = (tokbin[base + j] & TOKMASK) | (b6 << BINSHIFT);
  }
}

// ---------------- kernel 3: WMMA one-hot segment-sum + emb + KL ----------------
// Grid: (8 d-tiles, 64 batches); 256 threads = 8 waves, each wave owns S/8 = 256 s.
// Software-pipelined: tok/bin loaded 2 chunks ahead, W mu/lv 1 chunk ahead.
__global__ void __launch_bounds__(256)
k3_wmma_scatter(const float* __restrict__ W, const unsigned* __restrict__ tokbin,
                float* __restrict__ out, float* __restrict__ kl_out) {
  int dtile = blockIdx.x;          // 0..7  -> d range [dtile*16, dtile*16+16)
  int b     = blockIdx.y;          // 0..63
  int tid   = threadIdx.x;
  int wave  = tid >> 5, lane = tid & 31;
  int lhalf = lane & 15;           // A row m / B col n for this lane
  int khalf = (lane >> 4) << 1;    // lanes 0-15 -> K=0,1 ; lanes 16-31 -> K=2,3
  int dBase = dtile * 16;

  const int SPW = S_ / 8;          // 256 s-positions per wave
  const int NCH = SPW / 4;         // 64 K=4 chunks
  int sBase = wave * SPW;
  const unsigned* tbRow = tokbin + b * S_;

  v8f c0 = {}, c1 = {}, c2 = {};
  float klacc = 0.f;

  // -------- pipeline preamble --------
  // tb for chunk 0 (current) and chunk 1 (next); W values for chunk 0.
  int sC = sBase + khalf;
  unsigned tbA  = tbRow[sC];                 // chunk 0, slot k=khalf
  unsigned tbB  = tbRow[sC + 1];             // chunk 0, slot k=khalf+1
  unsigned ntbA = tbRow[sC + 4];             // chunk 1
  unsigned ntbB = tbRow[sC + 5];
  const float* rA = W + (size_t)(tbA & TOKMASK) * WROW + dBase + lhalf;
  const float* rB = W + (size_t)(tbB & TOKMASK) * WROW + dBase + lhalf;
  float muA = rA[0], lvA = rA[D_];
  float muB = rB[0], lvB = rB[D_];

  for (int chunk = 0; chunk < NCH; ++chunk) {
    int s0 = sBase + chunk * 4;
    int sA = s0 + khalf, sB = sA + 1;

    // ---- issue loads for chunk+2 (clamped; values never used past NCH-1) ----
    int cNext2 = (chunk + 2 < NCH) ? (chunk + 2) : (NCH - 1);
    int sF = sBase + cNext2 * 4 + khalf;
    unsigned ftbA = tbRow[sF];
    unsigned ftbB = tbRow[sF + 1];
    __builtin_prefetch(&tbRow[sBase + ((chunk + 8 < NCH) ? (chunk + 8) : (NCH - 1)) * 4], 0, 1);

    // ---- issue W loads for chunk+1 (tok already resident from last iter) ----
    const float* nrA = W + (size_t)(ntbA & TOKMASK) * WROW + dBase + lhalf;
    const float* nrB = W + (size_t)(ntbB & TOKMASK) * WROW + dBase + lhalf;
    float nmuA = nrA[0], nlvA = nrA[D_];
    float nmuB = nrB[0], nlvB = nrB[D_];

    // ---- compute current chunk ----
    unsigned binA = tbA >> BINSHIFT;
    unsigned binB = tbB >> BINSHIFT;

    unsigned iA = ((unsigned)(b * S_ + sA)) * D_ + (unsigned)(dBase + lhalf);
    unsigned iB = ((unsigned)(b * S_ + sB)) * D_ + (unsigned)(dBase + lhalf);
    float embA = fmaf(__expf(0.5f * lvA), hash_gauss(iA), muA);
    float embB = fmaf(__expf(0.5f * lvB), hash_gauss(iB), muB);

    klacc += (-2.0f - lvA - lvB + muA * muA + muB * muB + __expf(lvA) + __expf(lvB));

    // B tile (4 x 16): this lane supplies (K=khalf, n=lhalf) and (K=khalf+1, n=lhalf)
    v2f bv; bv.x = embA; bv.y = embB;
    // A tiles (16 x 4) one-hot over bins, m = lhalf, K slots mirror B
    v2f a0, a1, a2;
    a0.x = (binA == (unsigned)lhalf)        ? 1.f : 0.f;
    a0.y = (binB == (unsigned)lhalf)        ? 1.f : 0.f;
    a1.x = (binA == (unsigned)(16 + lhalf)) ? 1.f : 0.f;
    a1.y = (binB == (unsigned)(16 + lhalf)) ? 1.f : 0.f;
    a2.x = (binA == (unsigned)(32 + lhalf)) ? 1.f : 0.f;
    a2.y = (binB == (unsigned)(32 + lhalf)) ? 1.f : 0.f;

    c0 = __builtin_amdgcn_wmma_f32_16x16x4_f32(false, a0, false, bv, (short)0, c0, false, false);
    c1 = __builtin_amdgcn_wmma_f32_16x16x4_f32(false, a1, false, bv, (short)0, c1, false, false);
    c2 = __builtin_amdgcn_wmma_f32_16x16x4_f32(false, a2, false, bv, (short)0, c2, false, false);

    // ---- rotate pipeline ----
    tbA = ntbA; tbB = ntbB; ntbA = ftbA; ntbB = ftbB;
    muA = nmuA; lvA = nlvA; muB = nmuB; lvB = nlvB;
  }

  // ---- combine the 8 waves' partial C tiles in LDS (ds_add_f32) ----
  __shared__ float red[NBINS * 16];
  for (int i = tid; i < NBINS * 16; i += 256) red[i] = 0.f;
  __syncthreads();

  int mrow_off = (lane >= 16) ? 8 : 0;   // C layout: VGPR j, lanes>=16 hold M=j+8
#pragma unroll
  for (int j = 0; j < 8; ++j) {
    int m = j + mrow_off, n = lhalf;
    atomicAdd(&red[( 0 + m) * 16 + n], c0[j]);
    atomicAdd(&red[(16 + m) * 16 + n], c1[j]);
    atomicAdd(&red[(32 + m) * 16 + n], c2[j]);
  }
  __syncthreads();

  // exclusive (b, d-tile) ownership -> plain stores, no pre-zero of out needed
  for (int i = tid; i < NBINS * 16; i += 256) {
    int m = i >> 4, n = i & 15;
    out[((size_t)b * NBINS + m) * D_ + dBase + n] = red[i];
  }

  // ---- KL block reduction -> one global atomic per block ----
  __shared__ float kls[256];
  kls[tid] = klacc; __syncthreads();
  for (int off = 128; off > 0; off >>= 1) {
    if (tid < off) kls[tid] += kls[tid + off];
    __syncthreads();
  }
  if (tid == 0) atomicAdd(kl_out, kls[0] * (0.5f / (float)B_));
}

// ---------------- launcher ----------------
extern "C" void kernel_launch(void* const* d_in, const int* in_sizes, int n_in,
                              void* d_out, int out_size, void* d_ws, size_t ws_size,
                              hipStream_t stream) {
  const float* X = (const float*)d_in[0];   // (B, S, 2) f32
  const float* W = (const float*)d_in[1];   // (N_TOKENS, 256) f32
  float* out = (float*)d_out;               // (B,48,128) + KL scalar

  float*    H      = (float*)d_ws;                                      // B*S f32
  unsigned* tokbin = (unsigned*)((char*)d_ws + sizeof(float) * B_ * S_);// B*S u32
  float* klslot = out + (size_t)B_ * NBINS * D_;

  k1_entropy   <<<(B_ * S_ + 255) / 256, 256, 0, stream>>>(X, W, H, tokbin, klslot);
  k2_scan_bins <<<B_, 256, 0, stream>>>(H, tokbin);
  k3_wmma_scatter<<<dim3(8, B_), 256, 0, stream>>>(W, tokbin, out, klslot);
}